// RNNPropOptimizer_31593779429594
// MI455X (gfx1250) — compile-verified
//
#include <hip/hip_runtime.h>
#include <math.h>

typedef __attribute__((ext_vector_type(2))) float v2f;
typedef __attribute__((ext_vector_type(8))) float v8f;

#define HDIM 20
#define GDIM 80
#define ZSTR 84            // padded row stride for z scratch (avoids 2-way bank conflict at 80)
#define WAVES 4
#define BLOCK (WAVES * 32)

// LDS layout (float offsets). Weight matrices stored PAIR-ROW INTERLEAVED:
//   dest[(k>>1)*160 + n*2 + (k&1)] = M[k][n]
// so a WMMA B-fragment (M[k][n], M[k+1][n]) is one aligned ds_load_b64.
#define WOFF_W1P 0         // [2 pair-rows x 80 x 2], pair-row 1 zero-padded
#define WOFF_U1  320       // [10 x 80 x 2]
#define WOFF_B1  1920      // [80]
#define WOFF_W2  2000      // [10 x 80 x 2]
#define WOFF_U2  3600      // [10 x 80 x 2]
#define WOFF_B2  5200      // [80]
#define WOFF_WD  5280      // [20]
#define WOFF_BD  5300      // [1]
#define WOFF_END 5312      // 8-float aligned
#define PER_WAVE 2624      // zb(16*84=1344) + hA(320) + hB(320) + cA(320) + cB(320)
#define LDS_FLOATS (WOFF_END + WAVES * PER_WAVE)   // 15808 floats = 63232 B

#define LOG2E 1.4426950408889634f

static __device__ __forceinline__ v8f wmma_f32(v2f a, v2f b, v8f c) {
    // V_WMMA_F32_16X16X4_F32 : D = A(16x4) * B(4x16) + C(16x16), all f32
    return __builtin_amdgcn_wmma_f32_16x16x4_f32(
        /*neg_a=*/false, a, /*neg_b=*/false, b,
        /*c_mod=*/(short)0, c, /*reuse_a=*/false, /*reuse_b=*/false);
}

// hardware-transcendental tanh (v_exp_f32 + v_rcp_f32); saturates correctly for |x| large
static __device__ __forceinline__ float fast_tanh(float x) {
    return __builtin_amdgcn_rcpf(1.0f + __builtin_amdgcn_exp2f(x * (-2.0f * LOG2E))) * 2.0f - 1.0f;
}

// acc[5] += A[16x20](LDS, row stride 20) * B[20x80](LDS, pair-row interleaved)
static __device__ __forceinline__ void gemm_h(const float* A, const float* B,
                                              v8f acc[5], int lane) {
    const int mrow = lane & 15;
    const int koff = (lane >> 4) << 1;   // 0 for lanes 0-15, 2 for lanes 16-31
#pragma unroll
    for (int k0 = 0; k0 < HDIM; k0 += 4) {
        // A fragment: 8-byte aligned pair (mrow*20 even, k0+koff even)
        v2f a = *(const v2f*)(A + mrow * HDIM + k0 + koff);
        const int prow = (k0 + koff) >> 1;     // pair-row index
        v2f b[5];
#pragma unroll
        for (int nt = 0; nt < 5; ++nt)
            b[nt] = *(const v2f*)(B + prow * 160 + (nt * 16 + mrow) * 2);  // one ds_load_b64
#pragma unroll
        for (int nt = 0; nt < 5; ++nt)
            acc[nt] = wmma_f32(a, b[nt], acc[nt]);
    }
}

// bias + gate activations (branch-free tanh/sigmoid select); z[16x80] -> LDS (stride ZSTR)
static __device__ __forceinline__ void act_store(v8f acc[5], const float* bias,
                                                 float* zb, int lane) {
    const int lcol = lane & 15;
    const int mAdd = (lane >> 4) << 3;   // C layout: vgpr j -> row j (+8 for upper lanes)
#pragma unroll
    for (int nt = 0; nt < 5; ++nt) {
        const int col = nt * 16 + lcol;
        const float bv = bias[col];
        const bool isG = (col >= 40) && (col < 60);     // cell gate -> tanh, else sigmoid
        const float sc = isG ? (-2.0f * LOG2E) : -LOG2E;
        const float mu = isG ? 2.0f : 1.0f;
        const float ad = isG ? -1.0f : 0.0f;
#pragma unroll
        for (int j = 0; j < 8; ++j) {
            float zv = acc[nt][j] + bv;
            float e = __builtin_amdgcn_exp2f(zv * sc);
            float a = __builtin_amdgcn_rcpf(1.0f + e) * mu + ad;
            zb[(j + mAdd) * ZSTR + col] = a;
        }
    }
}

// c_new = f*c + i*g ; h_new = o*tanh(c_new). Two adjacent elements per iteration:
// pairs start even and rows are 20 wide, so every gate/c pair merges to b64,
// and global stores are 8-byte wide + coalesced.
static __device__ __forceinline__ void cell_update(const float* zb, const float* cbuf,
                                                   float* hdst,
                                                   float* __restrict__ gc,
                                                   float* __restrict__ gh, int lane) {
#pragma unroll
    for (int r = 0; r < 5; ++r) {
        const int e0 = (lane + 32 * r) * 2;   // 0..318 even, over [16 x 20] flat
        float cn[2], hn[2];
#pragma unroll
        for (int t = 0; t < 2; ++t) {
            const int e = e0 + t;
            const int M = e / HDIM;
            const int hc = e - M * HDIM;
            const float* zr = zb + M * ZSTR;
            float iv = zr[hc];
            float fv = zr[20 + hc];
            float gv = zr[40 + hc];
            float ov = zr[60 + hc];
            float c_ = fv * cbuf[e] + iv * gv;
            cn[t] = c_;
            hn[t] = ov * fast_tanh(c_);
        }
        v2f cw; cw.x = cn[0]; cw.y = cn[1];
        v2f hw; hw.x = hn[0]; hw.y = hn[1];
        *(v2f*)(hdst + e0) = hw;              // LDS b64: feed next stage
        *(v2f*)(gc + e0) = cw;                // global_store_b64, coalesced
        *(v2f*)(gh + e0) = hw;                // global_store_b64, coalesced
    }
}

__global__ __launch_bounds__(BLOCK) void rnnprop_kernel(
    const float* __restrict__ grads, const float* __restrict__ m_in, const float* __restrict__ v_in,
    const float* __restrict__ h1_in, const float* __restrict__ c1_in,
    const float* __restrict__ h2_in, const float* __restrict__ c2_in,
    const float* __restrict__ W1, const float* __restrict__ U1, const float* __restrict__ b1,
    const float* __restrict__ W2, const float* __restrict__ U2, const float* __restrict__ b2,
    const float* __restrict__ Wd, const float* __restrict__ bd,
    float* __restrict__ o_up, float* __restrict__ o_m, float* __restrict__ o_v,
    float* __restrict__ o_h1, float* __restrict__ o_c1,
    float* __restrict__ o_h2, float* __restrict__ o_c2,
    long long n)
{
    __shared__ float lds[LDS_FLOATS];
    const int tid = threadIdx.x;
    const int lane = tid & 31;
    const int wave = tid >> 5;

    // ---- cooperative weight load, scattering into pair-row interleaved layout ----
    for (int i = tid; i < 1600; i += BLOCK) {
        const int k = i / GDIM, c = i - k * GDIM;
        const int d = (k >> 1) * 160 + c * 2 + (k & 1);
        lds[WOFF_U1 + d] = U1[i];
        lds[WOFF_W2 + d] = W2[i];
        lds[WOFF_U2 + d] = U2[i];
    }
    for (int i = tid; i < 320; i += BLOCK) {           // W1 padded to 4 K-rows (2..3 zero)
        const int k = i / GDIM, c = i - k * GDIM;
        const int d = (k >> 1) * 160 + c * 2 + (k & 1);
        lds[WOFF_W1P + d] = (k < 2) ? W1[i] : 0.0f;
    }
    for (int i = tid; i < 80; i += BLOCK) {
        lds[WOFF_B1 + i] = b1[i];
        lds[WOFF_B2 + i] = b2[i];
    }
    if (tid < HDIM) lds[WOFF_WD + tid] = Wd[tid];
    if (tid == 0) lds[WOFF_BD] = bd[0];
    __syncthreads();

    const long long p0 = ((long long)blockIdx.x * WAVES + wave) * 16;
    if (p0 + 16 > n) return;   // no barriers after this point

    float* zb = lds + WOFF_END + wave * PER_WAVE;
    float* hA = zb + 16 * ZSTR;   // h1 -> h1n -> h2n
    float* hB = hA + 320;         // h2 (layer-2 recurrent input)
    float* cA = hB + 320;         // c1
    float* cB = cA + 320;         // c2

    // ---- elementwise rms_momentum (lanes 0-15 own one param each) ----
    float mt = 0.f, gt = 0.f;    // lanes >=16 represent K=2,3 of the A fragment -> zero
    if (lane < 16) {
        const long long p = p0 + lane;
        float g = grads[p];
        float mn = 0.9f * m_in[p] + 0.1f * g;
        float vn = 0.9f * v_in[p] + 0.1f * g * g;
        o_m[p] = mn;
        o_v[p] = vn;
        float inv = __builtin_amdgcn_rsqf(vn * 10.0f + 1e-10f);  // 1/sqrt(v_hat + eps)
        mt = (mn * 10.0f) * inv;                                 // m_tilde
        gt = g * inv;                                            // g_tilde
    }

    // ---- stage LSTM states with b64 loads, fully coalesced ----
    const long long s0 = p0 * HDIM;          // tile start, 8B-aligned (p0 multiple of 16)
    {
        const v2f* h1p = (const v2f*)(h1_in + s0);
        const v2f* c1p = (const v2f*)(c1_in + s0);
        const v2f* h2p = (const v2f*)(h2_in + s0);
        const v2f* c2p = (const v2f*)(c2_in + s0);
        v2f* hA2 = (v2f*)hA; v2f* cA2 = (v2f*)cA;
        v2f* hB2 = (v2f*)hB; v2f* cB2 = (v2f*)cB;
#pragma unroll
        for (int r = 0; r < 5; ++r) {
            const int e = lane + 32 * r;     // 0..159 float2 index
            hA2[e] = h1p[e];
            cA2[e] = c1p[e];
            hB2[e] = h2p[e];
            cB2[e] = c2p[e];
        }
    }

    v8f acc[5];
    const v8f zeroc = {0.f, 0.f, 0.f, 0.f, 0.f, 0.f, 0.f, 0.f};

    // ================= Layer 1 =================
#pragma unroll
    for (int nt = 0; nt < 5; ++nt) acc[nt] = zeroc;

    // input term: x[16x2] @ W1[2x80] via zero-padded interleaved W1 -> no divergence
    {
        v2f a; a.x = mt; a.y = gt;
        const int mrow = lane & 15;
        const int prow = lane >> 4;          // pair-row 0 (K=0,1) or 1 (zeros, K=2,3)
        v2f b[5];
#pragma unroll
        for (int nt = 0; nt < 5; ++nt)
            b[nt] = *(const v2f*)(lds + WOFF_W1P + prow * 160 + (nt * 16 + mrow) * 2);
#pragma unroll
        for (int nt = 0; nt < 5; ++nt)
            acc[nt] = wmma_f32(a, b[nt], acc[nt]);
    }
    gemm_h(hA, lds + WOFF_U1, acc, lane);            // + h1 @ U1
    act_store(acc, lds + WOFF_B1, zb, lane);
    cell_update(zb, cA, hA, o_c1 + s0, o_h1 + s0, lane);   // hA := h1n

    // ================= Layer 2 =================
#pragma unroll
    for (int nt = 0; nt < 5; ++nt) acc[nt] = zeroc;
    gemm_h(hA, lds + WOFF_W2, acc, lane);            // h1n @ W2
    gemm_h(hB, lds + WOFF_U2, acc, lane);            // + h2 @ U2
    act_store(acc, lds + WOFF_B2, zb, lane);
    cell_update(zb, cB, hA, o_c2 + s0, o_h2 + s0, lane);   // hA := h2n

    // ================= Decoder: update = 0.1 * tanh(h2n @ Wd + bd) =================
    if (lane < 16) {
        float s = lds[WOFF_BD];
        const v2f* hrow = (const v2f*)(hA + lane * HDIM);   // lane*20 even -> aligned
#pragma unroll
        for (int j = 0; j < HDIM / 2; ++j) {
            v2f hp = hrow[j];                                // ds_load_b64
            s += hp.x * lds[WOFF_WD + 2 * j] + hp.y * lds[WOFF_WD + 2 * j + 1];
        }
        o_up[p0 + lane] = 0.1f * fast_tanh(s);
    }
}

extern "C" void kernel_launch(void* const* d_in, const int* in_sizes, int n_in,
                              void* d_out, int out_size, void* d_ws, size_t ws_size,
                              hipStream_t stream) {
    // setup_inputs order: param, grads, m, v, h1, c1, h2, c2, W1, U1, b1, W2, U2, b2, Wd, bd
    const float* grads = (const float*)d_in[1];
    const float* m_in  = (const float*)d_in[2];
    const float* v_in  = (const float*)d_in[3];
    const float* h1    = (const float*)d_in[4];
    const float* c1    = (const float*)d_in[5];
    const float* h2    = (const float*)d_in[6];
    const float* c2    = (const float*)d_in[7];
    const float* W1    = (const float*)d_in[8];
    const float* U1    = (const float*)d_in[9];
    const float* b1    = (const float*)d_in[10];
    const float* W2    = (const float*)d_in[11];
    const float* U2    = (const float*)d_in[12];
    const float* b2    = (const float*)d_in[13];
    const float* Wd    = (const float*)d_in[14];
    const float* bd    = (const float*)d_in[15];

    const long long N = (long long)in_sizes[1];
    float* out = (float*)d_out;
    // outputs concatenated flat: update[N], m[N], v[N], h1n[N*20], c1n[N*20], h2n[N*20], c2n[N*20]
    float* o_up = out;
    float* o_m  = out + N;
    float* o_v  = out + 2 * N;
    float* o_h1 = out + 3 * N;
    float* o_c1 = out + 23 * N;
    float* o_h2 = out + 43 * N;
    float* o_c2 = out + 63 * N;

    const int blocks = (int)((N + (WAVES * 16) - 1) / (WAVES * 16));
    rnnprop_kernel<<<blocks, BLOCK, 0, stream>>>(
        grads, m_in, v_in, h1, c1, h2, c2,
        W1, U1, b1, W2, U2, b2, Wd, bd,
        o_up, o_m, o_v, o_h1, o_c1, o_h2, o_c2, N);
}